// NeumannCayleySSM_62302795596606
// MI455X (gfx1250) — compile-verified
//
#include <hip/hip_runtime.h>
#include <math.h>

// ---------------------------------------------------------------------------
// NeumannCayley SSM for MI455X (gfx1250, wave32).
// fp32 end-to-end using V_WMMA_F32_16X16X4_F32 (n=16 state => native tile).
// Sequential recurrence parallelized via two-level affine-map scan.
// ---------------------------------------------------------------------------

#define BATCHN   8
#define SEQL     4096
#define DMODEL   1024
#define TOKENS   (BATCHN * SEQL)      // 32768
#define NST      16
#define NP       120                  // skew params
#define NCOLS    136                  // 120 skew + 16 Bx
#define RHO_MAX  0.3f
#define CHUNK    64
#define NCHUNK   (SEQL / CHUNK)       // 64
#define NCHAINS  (BATCHN * NCHUNK)    // 512
#define TP       17                   // LDS pitch for 16x16 tiles (bank-conflict free)
#define KC       64                   // K-chunk for the big projection GEMM

typedef __attribute__((ext_vector_type(2))) float v2f;
typedef __attribute__((ext_vector_type(8))) float v8f;

__device__ __forceinline__ v8f wmma4(v2f a, v2f b, v8f c) {
  // D = A(16x4,f32) x B(4x16,f32) + C(16x16,f32)
  return __builtin_amdgcn_wmma_f32_16x16x4_f32(false, a, false, b, (short)0, c,
                                               false, false);
}
__device__ __forceinline__ v8f zero8() {
  v8f z = {0.f, 0.f, 0.f, 0.f, 0.f, 0.f, 0.f, 0.f};
  return z;
}
__device__ __forceinline__ float wave_sum(float x) {
#pragma unroll
  for (int off = 16; off > 0; off >>= 1) x += __shfl_xor(x, off, 32);
  return x;
}
// C/D layout: VGPR r -> element (M = r + 8*(lane>=16), N = lane&15)
__device__ __forceinline__ void store_cd(float* T, int lane, v8f acc) {
  int n = lane & 15, mb = (lane >> 4) << 3;
#pragma unroll
  for (int r = 0; r < 8; ++r) T[(mb + r) * TP + n] = acc[r];
}
__device__ __forceinline__ int triidx(int r, int c) {
  return r * 15 - (r * (r - 1)) / 2 + (c - r - 1);  // triu_indices(16,1) flat idx
}
__device__ __forceinline__ unsigned hashu(unsigned x) {
  x ^= x >> 16; x *= 0x7feb352dU; x ^= x >> 15; x *= 0x846ca68bU; x ^= x >> 16;
  return x;
}
__device__ __forceinline__ float rnd_normal(unsigned token, unsigned i) {
  unsigned s1 = hashu(token * 0x9e3779b9U + i * 0x85ebca6bU + 0x165667b1U);
  unsigned s2 = hashu(s1 + 0x27d4eb2fU);
  float u1 = ((float)(s1 >> 8) + 1.0f) * (1.0f / 16777217.0f);
  float u2 = (float)(s2 >> 8) * (1.0f / 16777216.0f);
  return sqrtf(-2.0f * logf(u1)) * cosf(6.28318530718f * u2);
}

// ---------------------------------------------------------------------------
// Kernel 1: vbx[token][0:120] = u @ Wpᵀ + b ; vbx[token][120:136] = u @ Bwᵀ
// Block = 288 threads (9 waves). Each block: one 16-token M-tile; wave w
// owns output columns [16w, 16w+16). K staged through LDS in chunks of 64.
// ---------------------------------------------------------------------------
__global__ __launch_bounds__(288) void proj_kernel(const float* __restrict__ u,
                                                   const float* __restrict__ Wp,
                                                   const float* __restrict__ bp,
                                                   const float* __restrict__ Bw,
                                                   float* __restrict__ vbx) {
  __shared__ float su[16][KC + 4];    // pitch 68: conflict-free column reads
  __shared__ float sw[144][KC + 4];
  const int tid = threadIdx.x;
  const int lane = tid & 31, wid = tid >> 5;  // wid 0..8
  const int m0 = blockIdx.x * 16;
  const int m16 = lane & 15;
  const int hi = (lane >> 4) << 1;            // 0 for lanes 0-15, 2 for 16-31
  v8f acc = zero8();

  for (int kc = 0; kc < DMODEL; kc += KC) {
    for (int i = tid; i < 16 * KC; i += 288) {
      int r = i / KC, c = i % KC;
      su[r][c] = u[(size_t)(m0 + r) * DMODEL + kc + c];
    }
    for (int i = tid; i < 144 * KC; i += 288) {
      int r = i / KC, c = i % KC;
      float w = 0.f;
      if (r < NP)         w = Wp[(size_t)r * DMODEL + kc + c];
      else if (r < NCOLS) w = Bw[(size_t)(r - NP) * DMODEL + kc + c];
      sw[r][c] = w;
    }
    __syncthreads();
    const int nb = wid * 16 + m16;  // weight row == output column
#pragma unroll
    for (int s = 0; s < KC; s += 4) {
      v2f a, b;
      a.x = su[m16][s + hi];  a.y = su[m16][s + hi + 1];
      b.x = sw[nb][s + hi];   b.y = sw[nb][s + hi + 1];
      acc = wmma4(a, b, acc);
    }
    __syncthreads();
  }
  const int col = wid * 16 + m16;
  const int mb = (lane >> 4) << 3;
  if (col < NCOLS) {
    const float bias = (col < NP) ? bp[col] : 0.f;
#pragma unroll
    for (int r = 0; r < 8; ++r)
      vbx[(size_t)(m0 + mb + r) * NCOLS + col] = acc[r] + bias;
  }
}

// ---------------------------------------------------------------------------
// Kernel 2: per token -- build skew A, power-iterate sigma, scale,
// W = (I-A)(I+A^2)(I-A) via WMMA, ||WᵀW - I||_F accumulated for ortho_dev.
// One wave per token, 8 waves/block, per-wave private LDS tiles.
// ---------------------------------------------------------------------------
__global__ __launch_bounds__(256) void build_w_kernel(const float* __restrict__ vbx,
                                                      float* __restrict__ Wout,
                                                      float* __restrict__ ortho_acc) {
  __shared__ float tiles[8][2][16 * TP];
  const int lane = threadIdx.x & 31, w = threadIdx.x >> 5;
  const int token = blockIdx.x * 8 + w;
  float* At = &tiles[w][0][0];
  float* Tt = &tiles[w][1][0];
  const float* vp = vbx + (size_t)token * NCOLS;
  const int m16 = lane & 15;
  const int hi = (lane >> 4) << 1;
  const int mb = (lane >> 4) << 3;

  // Build skew-symmetric A in LDS (each lane fills 8 entries).
  for (int e = lane; e < 256; e += 32) {
    int r = e >> 4, c = e & 15;
    float val = 0.f;
    if (r < c)      val =  vp[triidx(r, c)];
    else if (r > c) val = -vp[triidx(c, r)];
    At[r * TP + c] = val;
  }
  // (same-wave LDS ops are in-order; no cross-wave sharing)

  // Power iteration (2 iters).  sigma = ||A v|| since u = normalize(A v).
  float uv = (lane < 16) ? rnd_normal((unsigned)token, (unsigned)lane) : 0.f;
  {
    float n0 = sqrtf(wave_sum(uv * uv));
    uv = uv / fmaxf(n0, 1e-12f);
  }
  float sigma = 0.f;
#pragma unroll
  for (int it = 0; it < 2; ++it) {
    float s1 = 0.f;                        // (A u)[m16] on all lanes (halves dup)
#pragma unroll
    for (int j = 0; j < 16; ++j) s1 += At[m16 * TP + j] * __shfl(uv, j, 32);
    float w1 = -s1;                        // Aᵀu = -Au
    float w1m = (lane < 16) ? w1 : 0.f;
    float n1 = sqrtf(wave_sum(w1m * w1m));
    float vv = w1 / fmaxf(n1, 1e-12f);
    float s2 = 0.f;                        // z = A v
#pragma unroll
    for (int j = 0; j < 16; ++j) s2 += At[m16 * TP + j] * __shfl(vv, j, 32);
    float zm = (lane < 16) ? s2 : 0.f;
    float n2 = sqrtf(wave_sum(zm * zm));
    uv = s2 / fmaxf(n2, 1e-12f);
    sigma = n2;
  }
  const float scale = RHO_MAX / fmaxf(sigma, RHO_MAX);
  for (int e = lane; e < 256; e += 32) {
    int r = e >> 4, c = e & 15;
    At[r * TP + c] *= scale;
  }

  // T = I + A @ A
  v8f acc0 = zero8();
#pragma unroll
  for (int k0 = 0; k0 < 16; k0 += 4) {
    int ka = k0 + hi;
    v2f a, b;
    a.x = At[m16 * TP + ka];      a.y = At[m16 * TP + ka + 1];
    b.x = At[ka * TP + m16];      b.y = At[(ka + 1) * TP + m16];
    acc0 = wmma4(a, b, acc0);
  }
  const int n = m16;
#pragma unroll
  for (int r = 0; r < 8; ++r)
    if (mb + r == n) acc0[r] += 1.f;
  store_cd(Tt, lane, acc0);

  // M1 = (I - A) @ T   (overwrite Tt with M1 after reads)
  v8f acc1 = zero8();
#pragma unroll
  for (int k0 = 0; k0 < 16; k0 += 4) {
    int ka = k0 + hi;
    v2f a, b;
    a.x = ((m16 == ka)     ? 1.f : 0.f) - At[m16 * TP + ka];
    a.y = ((m16 == ka + 1) ? 1.f : 0.f) - At[m16 * TP + ka + 1];
    b.x = Tt[ka * TP + m16];
    b.y = Tt[(ka + 1) * TP + m16];
    acc1 = wmma4(a, b, acc1);
  }
  store_cd(Tt, lane, acc1);

  // W = M1 @ (I - A)
  v8f Wacc = zero8();
#pragma unroll
  for (int k0 = 0; k0 < 16; k0 += 4) {
    int ka = k0 + hi;
    v2f a, b;
    a.x = Tt[m16 * TP + ka];      a.y = Tt[m16 * TP + ka + 1];
    b.x = ((ka == m16)     ? 1.f : 0.f) - At[ka * TP + m16];
    b.y = ((ka + 1 == m16) ? 1.f : 0.f) - At[(ka + 1) * TP + m16];
    Wacc = wmma4(a, b, Wacc);
  }
  // Store W (global, row-major 16x16 per token) + stash in LDS for WᵀW.
#pragma unroll
  for (int r = 0; r < 8; ++r)
    Wout[(size_t)token * 256 + (mb + r) * 16 + n] = Wacc[r];
  store_cd(At, lane, Wacc);  // A no longer needed

  // WtW = Wᵀ W ;  A-frag (m,k)=W[k][m], B-frag (k,n)=W[k][n]: identical (m==n).
  v8f acc2 = zero8();
#pragma unroll
  for (int k0 = 0; k0 < 16; k0 += 4) {
    int ka = k0 + hi;
    v2f f;
    f.x = At[ka * TP + m16];
    f.y = At[(ka + 1) * TP + m16];
    acc2 = wmma4(f, f, acc2);
  }
  float ss = 0.f;
#pragma unroll
  for (int r = 0; r < 8; ++r) {
    float d = acc2[r] - ((mb + r == n) ? 1.f : 0.f);
    ss += d * d;
  }
  float tot = wave_sum(ss);
  if (lane == 0) atomicAdd(ortho_acc, sqrtf(tot));
}

// ---------------------------------------------------------------------------
// Kernel 3a: per (batch,chunk) compose affine maps over 64 steps:
//   q <- W_t q + bx_t (shuffle matvec),  M <- W_t @ M (4 WMMAs, M in LDS).
// One wave per chunk; 512 chunks total.
// ---------------------------------------------------------------------------
__global__ __launch_bounds__(256) void chunk_compose_kernel(const float* __restrict__ W,
                                                            const float* __restrict__ vbx,
                                                            float* __restrict__ Pbuf,
                                                            float* __restrict__ qbuf) {
  __shared__ float mt[8][16 * TP];
  const int lane = threadIdx.x & 31, w = threadIdx.x >> 5;
  const int chain = blockIdx.x * 8 + w;         // 0..511
  const int b = chain / NCHUNK, c = chain % NCHUNK;
  float* Mt = &mt[w][0];
  const int m16 = lane & 15;
  const bool hiHalf = lane >= 16;
  const int mb = (lane >> 4) << 3;

  for (int e = lane; e < 256; e += 32) {        // M = I
    int r = e >> 4, cc = e & 15;
    Mt[r * TP + cc] = (r == cc) ? 1.f : 0.f;
  }
  float q = 0.f;
  v8f acc = zero8();
  for (int s = 0; s < CHUNK; ++s) {
    const size_t token = (size_t)b * SEQL + (size_t)c * CHUNK + s;
    const float* Wt = W + token * 256;
    __builtin_prefetch(Wt + 256, 0, 1);         // -> global_prefetch_b8
    float rowv[16];
    const float4* rp = reinterpret_cast<const float4*>(Wt + m16 * 16);
#pragma unroll
    for (int i = 0; i < 4; ++i) {
      float4 t4 = rp[i];
      rowv[4 * i + 0] = t4.x; rowv[4 * i + 1] = t4.y;
      rowv[4 * i + 2] = t4.z; rowv[4 * i + 3] = t4.w;
    }
    float nq = 0.f;
#pragma unroll
    for (int j = 0; j < 16; ++j) nq += rowv[j] * __shfl(q, j, 32);
    q = nq + vbx[token * NCOLS + NP + m16];

    acc = zero8();
#pragma unroll
    for (int k0 = 0; k0 < 16; k0 += 4) {
      int ka = k0 + (hiHalf ? 2 : 0);
      v2f a, bb;
      a.x = hiHalf ? rowv[k0 + 2] : rowv[k0];
      a.y = hiHalf ? rowv[k0 + 3] : rowv[k0 + 1];
      bb.x = Mt[ka * TP + m16];
      bb.y = Mt[(ka + 1) * TP + m16];
      acc = wmma4(a, bb, acc);
    }
    store_cd(Mt, lane, acc);
  }
#pragma unroll
  for (int r = 0; r < 8; ++r)
    Pbuf[(size_t)chain * 256 + (mb + r) * 16 + m16] = acc[r];
  if (lane < 16) qbuf[(size_t)chain * 16 + m16] = q;
}

// Kernel 3b: sequential scan over NCHUNK chunk-transforms per batch (tiny).
__global__ __launch_bounds__(256) void chunk_scan_kernel(const float* __restrict__ Pbuf,
                                                         const float* __restrict__ qbuf,
                                                         float* __restrict__ Hstart) {
  const int lane = threadIdx.x & 31, b = threadIdx.x >> 5;  // 8 waves = 8 batches
  const int m16 = lane & 15;
  float h = 0.f;
  for (int c = 0; c < NCHUNK; ++c) {
    const int chain = b * NCHUNK + c;
    if (lane < 16) Hstart[(size_t)chain * 16 + m16] = h;
    const float* P = Pbuf + (size_t)chain * 256;
    float rowv[16];
    const float4* rp = reinterpret_cast<const float4*>(P + m16 * 16);
#pragma unroll
    for (int i = 0; i < 4; ++i) {
      float4 t4 = rp[i];
      rowv[4 * i + 0] = t4.x; rowv[4 * i + 1] = t4.y;
      rowv[4 * i + 2] = t4.z; rowv[4 * i + 3] = t4.w;
    }
    float nh = 0.f;
#pragma unroll
    for (int j = 0; j < 16; ++j) nh += rowv[j] * __shfl(h, j, 32);
    h = nh + qbuf[(size_t)chain * 16 + m16];
  }
}

// Kernel 3c: replay each chunk from its known starting state, emit h_all.
__global__ __launch_bounds__(256) void recur_kernel(const float* __restrict__ W,
                                                    const float* __restrict__ vbx,
                                                    const float* __restrict__ Hstart,
                                                    float* __restrict__ h_all) {
  const int lane = threadIdx.x & 31, w = threadIdx.x >> 5;
  const int chain = blockIdx.x * 8 + w;
  const int b = chain / NCHUNK, c = chain % NCHUNK;
  const int m16 = lane & 15;
  float h = Hstart[(size_t)chain * 16 + m16];
  for (int s = 0; s < CHUNK; ++s) {
    const size_t token = (size_t)b * SEQL + (size_t)c * CHUNK + s;
    const float* Wt = W + token * 256;
    __builtin_prefetch(Wt + 256, 0, 1);
    float rowv[16];
    const float4* rp = reinterpret_cast<const float4*>(Wt + m16 * 16);
#pragma unroll
    for (int i = 0; i < 4; ++i) {
      float4 t4 = rp[i];
      rowv[4 * i + 0] = t4.x; rowv[4 * i + 1] = t4.y;
      rowv[4 * i + 2] = t4.z; rowv[4 * i + 3] = t4.w;
    }
    float nh = 0.f;
#pragma unroll
    for (int j = 0; j < 16; ++j) nh += rowv[j] * __shfl(h, j, 32);
    h = nh + vbx[token * NCOLS + NP + m16];
    if (lane < 16) h_all[token * 16 + m16] = h;
  }
}

// ---------------------------------------------------------------------------
// Kernel 4: y = h_all @ C_wᵀ + u * D.  K=16 -> 4 WMMAs per 16x16 tile.
// ---------------------------------------------------------------------------
__global__ __launch_bounds__(256) void out_kernel(const float* __restrict__ h_all,
                                                  const float* __restrict__ Cw,
                                                  const float* __restrict__ u,
                                                  const float* __restrict__ Dv,
                                                  float* __restrict__ y) {
  const int lane = threadIdx.x & 31, w = threadIdx.x >> 5;
  const int wave = blockIdx.x * 8 + w;          // 0..131071
  const int mt = wave >> 6, nt = wave & 63;     // 2048 x 64 tiles
  const int m0 = mt * 16, n0 = nt * 16;
  const int m16 = lane & 15;
  const int hi = (lane >> 4) << 1;
  const int mb = (lane >> 4) << 3;
  const float* hrow = h_all + (size_t)(m0 + m16) * 16;   // A rows
  const float* crow = Cw + (size_t)(n0 + m16) * 16;      // B[k][n]=Cw[n][k]
  v8f acc = zero8();
#pragma unroll
  for (int k0 = 0; k0 < 16; k0 += 4) {
    v2f a, b;
    a.x = hrow[k0 + hi]; a.y = hrow[k0 + hi + 1];
    b.x = crow[k0 + hi]; b.y = crow[k0 + hi + 1];
    acc = wmma4(a, b, acc);
  }
  const int col = n0 + m16;
  const float dcol = Dv[col];
#pragma unroll
  for (int r = 0; r < 8; ++r) {
    const size_t off = (size_t)(m0 + mb + r) * DMODEL + col;
    y[off] = acc[r] + u[off] * dcol;
  }
}

__global__ void finalize_kernel(const float* __restrict__ acc, float* __restrict__ out) {
  if (threadIdx.x == 0) out[0] = acc[0] * (1.0f / (float)TOKENS);
}

// ---------------------------------------------------------------------------
extern "C" void kernel_launch(void* const* d_in, const int* in_sizes, int n_in,
                              void* d_out, int out_size, void* d_ws, size_t ws_size,
                              hipStream_t stream) {
  (void)in_sizes; (void)n_in; (void)out_size; (void)ws_size;
  const float* u  = (const float*)d_in[0];   // (8,4096,1024)
  const float* Wp = (const float*)d_in[1];   // (120,1024)
  const float* bp = (const float*)d_in[2];   // (120,)
  const float* Bw = (const float*)d_in[3];   // (16,1024)
  const float* Cw = (const float*)d_in[4];   // (1024,16)
  const float* Dv = (const float*)d_in[5];   // (1024,)
  float* y  = (float*)d_out;                 // 33554432 y + 1 ortho_dev
  float* ws = (float*)d_ws;

  // Workspace layout (floats); total ~54 MB.
  float* vbx    = ws;                                   // TOKENS*136
  float* Wbuf   = vbx    + (size_t)TOKENS * NCOLS;      // TOKENS*256
  float* h_all  = Wbuf   + (size_t)TOKENS * 256;        // TOKENS*16
  float* Pbuf   = h_all  + (size_t)TOKENS * 16;         // 512*256
  float* qbuf   = Pbuf   + (size_t)NCHAINS * 256;       // 512*16
  float* Hstart = qbuf   + (size_t)NCHAINS * 16;        // 512*16
  float* oacc   = Hstart + (size_t)NCHAINS * 16;        // 1

  hipMemsetAsync(oacc, 0, sizeof(float), stream);

  proj_kernel<<<TOKENS / 16, 288, 0, stream>>>(u, Wp, bp, Bw, vbx);
  build_w_kernel<<<TOKENS / 8, 256, 0, stream>>>(vbx, Wbuf, oacc);
  chunk_compose_kernel<<<NCHAINS / 8, 256, 0, stream>>>(Wbuf, vbx, Pbuf, qbuf);
  chunk_scan_kernel<<<1, 256, 0, stream>>>(Pbuf, qbuf, Hstart);
  recur_kernel<<<NCHAINS / 8, 256, 0, stream>>>(Wbuf, vbx, Hstart, h_all);
  out_kernel<<<(TOKENS / 16) * (DMODEL / 16) / 8, 256, 0, stream>>>(h_all, Cw, u, Dv, y);
  finalize_kernel<<<1, 32, 0, stream>>>(oacc, y + (size_t)TOKENS * DMODEL);
}